// KerasMultiLIFLayerSparse_46428596470115
// MI455X (gfx1250) — compile-verified
//
#include <hip/hip_runtime.h>
#include <cstdint>

// ---------------------------------------------------------------------------
// 2-layer LIF SNN scan on MI455X (gfx1250, wave32).
//  - 4 groups x 16 WGs (cluster-sized), 256 thr/WG, persistent over 512 steps
//  - membrane state in registers; spikes exchanged via L2 staging + barrier
//  - layer-2: V_WMMA_F32_16X16X4_F32 with A from spike-bit LUT (LDS @ offset 0)
//    and B from an LDS-resident 256KB W2 row-slice staged once by the TDM
// ---------------------------------------------------------------------------

#define SEQ     512
#define BATCH   64
#define NN      1024      // N_IN = N1 = N2
#define SMAX    64
#define GROUPS  4
#define WGPG    16        // workgroups per group (== cluster max)
#define CHUNK   64        // neurons owned per WG per layer
#define THREADS 256

typedef __attribute__((ext_vector_type(2))) float    v2f;
typedef __attribute__((ext_vector_type(4))) float    v4f;
typedef __attribute__((ext_vector_type(8))) float    v8f;
typedef __attribute__((ext_vector_type(4))) unsigned v4u;
typedef __attribute__((ext_vector_type(4))) int      v4i;
typedef __attribute__((ext_vector_type(8))) int      v8i;

#if defined(__has_builtin)
# if __has_builtin(__builtin_amdgcn_tensor_load_to_lds) && \
     __has_builtin(__builtin_amdgcn_s_wait_tensorcnt)
#  define HAVE_TDM 1
# endif
#endif

// ---- output layout (flat element offsets, tuple return order) -------------
#define OFF_IDS1 0
#define OFF_IDS2 (SEQ*BATCH*SMAX)
#define OFF_CNT1 (2*SEQ*BATCH*SMAX)
#define OFF_CNT2 (OFF_CNT1 + SEQ*BATCH*2)
#define OFF_ST1  (OFF_CNT2 + SEQ*BATCH*2)
#define OFF_ST2  (OFF_ST1  + (size_t)SEQ*BATCH*NN)

// ---- workspace layout (bytes) ----------------------------------------------
#define WS_W1T   0u
#define WS_STG1  ((size_t)NN*NN*4)
#define WS_STG2  (WS_STG1 + (size_t)BATCH*32*4)
#define WS_BAR   (WS_STG2 + (size_t)BATCH*32*4)

// ---- dynamic LDS layout (bytes); LUT at 0 so DS offsets need no base add ---
#define L_LUT   0u                           // [8] f32 (pair -> float2)
#define L_W2    32u                          // [64][1024] f32   (256 KB)
#define L_IDS   (L_W2  + (size_t)CHUNK*NN*4) // [16][64] int
#define L_M1    (L_IDS + 16*SMAX*4)          // [512] u32
#define L_ACC   (L_M1  + 512*4)              // [4][32][8] f32
#define L_LM1   (L_ACC + 4*32*8*4)           // [16][2] u32
#define L_LM2   (L_LM1 + 128)                // [16][2] u32
#define SMEM_BYTES (L_LM2 + 128)

// device-scope (agent) load: bypass possibly-stale WGP$ lines
__device__ __forceinline__ unsigned ld_dev(const unsigned* p) {
  return __hip_atomic_load(p, __ATOMIC_RELAXED, __HIP_MEMORY_SCOPE_AGENT);
}

// inter-workgroup barrier for the 16 WGs of one group.
// Emits the CDNA5 cluster barrier (S_NOP when not cluster-dispatched) plus a
// device-scope atomic barrier that is correct under any dispatch mode.
__device__ __forceinline__ unsigned group_barrier(unsigned* ctr, unsigned epoch) {
  __syncthreads();
  ++epoch;                                    // uniform across the block
  if (threadIdx.x == 0) {
    __threadfence();                          // release our staging stores
    __builtin_amdgcn_s_cluster_barrier();     // s_barrier_signal/wait -3
    __hip_atomic_fetch_add(ctr, 1u, __ATOMIC_RELEASE, __HIP_MEMORY_SCOPE_AGENT);
    while (__hip_atomic_load(ctr, __ATOMIC_ACQUIRE, __HIP_MEMORY_SCOPE_AGENT)
           < epoch * WGPG)
      __builtin_amdgcn_s_sleep(2);
    __threadfence();                          // acquire
  }
  __syncthreads();
  return epoch;
}

// wave32 spike-list compaction == jax.lax.top_k on (2*spike - idx/n):
// spiking indices ascending, padded with non-spiking indices ascending.
__device__ void emit_ids(const unsigned* __restrict__ src,
                         float* __restrict__ dst, int lane) {
  unsigned wv = ld_dev(src + lane);           // 32 words x 32 bits = 1024 bits
  int pc   = __popc(wv);
  int incl = pc;
  for (int o = 1; o < 32; o <<= 1) {
    int n = __shfl_up(incl, o, 32);
    if (lane >= o) incl += n;
  }
  int excl  = incl - pc;
  int total = __shfl(incl, 31, 32);
  unsigned m = wv; int r = excl;
  while (m) {
    int b = __builtin_ctz(m); m &= m - 1;
    if (r < SMAX) dst[r] = (float)(lane * 32 + b);
    ++r;
  }
  if (total < SMAX) {
    unsigned cm = ~wv; int cr = lane * 32 - excl;
    while (cm) {
      int b = __builtin_ctz(cm); cm &= cm - 1;
      int slot = total + cr;
      if (slot >= SMAX) break;
      dst[slot] = (float)(lane * 32 + b);
      ++cr;
    }
  }
}

// ---------------------------------------------------------------------------
__global__ void init_k(int* __restrict__ cnt, int ncnt,
                       unsigned* __restrict__ bar, int nbar) {
  int i = blockIdx.x * blockDim.x + threadIdx.x;
  if (i < ncnt) cnt[i] = 0;
  if (i < nbar) bar[i] = 0u;
}

// dst[i*NN + n] = src[n*NN + i]   (1024x1024 tiled transpose, for W1 only)
__global__ void transpose_k(const float* __restrict__ src,
                            float* __restrict__ dst) {
  __shared__ float tile[32][33];
  int bx = blockIdx.x & 31, by = blockIdx.x >> 5;
  int tx = threadIdx.x;
  for (int j = threadIdx.y; j < 32; j += 8)
    tile[j][tx] = src[(size_t)(by * 32 + j) * NN + bx * 32 + tx];
  __syncthreads();
  for (int j = threadIdx.y; j < 32; j += 8)
    dst[(size_t)(bx * 32 + j) * NN + by * 32 + tx] = tile[tx][j];
}

// ---------------------------------------------------------------------------
__global__ __launch_bounds__(THREADS)
void lif_persistent(const float* __restrict__ dec1g, const float* __restrict__ dec2g,
                    const float* __restrict__ th1g,  const float* __restrict__ th2g,
                    const float* __restrict__ init0, const float* __restrict__ init1,
                    const int*   __restrict__ inp_ids, const int* __restrict__ inp_num,
                    const float* __restrict__ W1T,   const float* __restrict__ W2,
                    unsigned* __restrict__ stg1, unsigned* __restrict__ stg2,
                    unsigned* __restrict__ barctr, float* __restrict__ out) {
  const int tid  = threadIdx.x;
  const int g    = blockIdx.x >> 4;       // group (16 batches)
  const int w    = blockIdx.x & 15;       // 64-neuron chunk index
  const int lane = tid & 31;
  const int wave = tid >> 5;

  extern __shared__ char smem[];
  float     *s_lut         = (float*)         (smem + L_LUT);  // @0: zero-base DS
  float    (*s_w2)[NN]     = (float(*)[NN])   (smem + L_W2);   // W2 rows [64][1024]
  int      (*s_ids)[SMAX]  = (int(*)[SMAX])   (smem + L_IDS);
  unsigned  *s_m1          = (unsigned*)      (smem + L_M1);
  float    (*s_acc)[32][8] = (float(*)[32][8])(smem + L_ACC);
  unsigned (*s_lm1)[2]     = (unsigned(*)[2]) (smem + L_LM1);
  unsigned (*s_lm2)[2]     = (unsigned(*)[2]) (smem + L_LM2);

  if (tid < 8) { int p = tid >> 1; s_lut[tid] = (float)((tid & 1) ? (p >> 1) & 1 : p & 1); }

  // ---- one-time stage of this WG's B panel: W2 rows [64*w, 64*w+64) -------
  const float* w2src = W2 + (size_t)(w * CHUNK) * NN;
#ifdef HAVE_TDM
  if (tid == 0) {
    unsigned           ldsb = (unsigned)(uintptr_t)(void*)s_w2;       // LDS byte addr
    unsigned long long ga   = (unsigned long long)(uintptr_t)w2src;
    v4u G0;
    G0.x = 1u;                                           // count=1 (valid user D#)
    G0.y = ldsb;                                         // lds_addr
    G0.z = (unsigned)ga;                                 // global_addr[31:0]
    G0.w = (unsigned)((ga >> 32) & 0x01FFFFFFull) | (2u << 30);  // [56:32] | type=2
    v8i G1;
    G1[0] = (int)(2u << 16);                             // data_size=2 (4 bytes)
    G1[1] = (int)((NN & 0xFFFFu) << 16);                 // tensor_dim0 = 1024
    G1[2] = (int)((CHUNK & 0xFFFFu) << 16);              // tensor_dim1 = 64
    G1[3] = (int)((NN & 0xFFFFu) << 16);                 // tile_dim0 = 1024
    G1[4] = (int)(CHUNK & 0xFFFFu);                      // tile_dim1 = 64
    G1[5] = (int)NN;                                     // tensor_dim0_stride = 1024
    G1[6] = 0; G1[7] = 0;
    v4i Z = {0, 0, 0, 0};
# if __clang_major__ >= 23
    v8i Z8 = {0, 0, 0, 0, 0, 0, 0, 0};
    __builtin_amdgcn_tensor_load_to_lds(G0, G1, Z, Z, Z8, 0);
# else
    __builtin_amdgcn_tensor_load_to_lds(G0, G1, Z, Z, 0);
# endif
    __builtin_amdgcn_s_wait_tensorcnt(0);
  }
#else
  for (int i = tid; i < CHUNK * NN / 4; i += THREADS)
    ((v4f*)s_w2)[i] = ((const v4f*)w2src)[i];
#endif
  __syncthreads();

  // ---- layer-1 ownership: (batch bl, 4 consecutive neurons) ----
  const int bl  = tid >> 4;               // 0..15 local batch
  const int l16 = tid & 15;
  const int b1  = g * 16 + bl;            // global batch
  const int n1  = w * CHUNK + l16 * 4;    // first owned layer-1 neuron
  float v1r[4], d1[4], t1[4];
  for (int j = 0; j < 4; ++j) {
    v1r[j] = init0[(size_t)b1 * NN + n1 + j];
    d1[j]  = dec1g[n1 + j];
    t1[j]  = th1g[n1 + j];
  }

  // ---- layer-2 ownership (waves 0..3): one n2 column, 8 batches ----
  const int tile = wave & 3;
  const int cl   = tile * 16 + (lane & 15);   // local col 0..63
  const int col  = w * CHUNK + cl;            // global n2
  const int mhi  = (lane >> 4) * 8;
  float v2r[8], d2 = 0.f, t2 = 0.f;
  if (wave < 4) {
    d2 = dec2g[col]; t2 = th2g[col];
    for (int r = 0; r < 8; ++r)
      v2r[r] = init1[(size_t)(g * 16 + r + mhi) * NN + col];
  }

  unsigned* ctr = barctr + g * 16;        // 64B-spaced counters
  unsigned  epoch = 0;

  int*   cnt1 = (int*)out + OFF_CNT1;
  int*   cnt2 = (int*)out + OFF_CNT2;
  float* ids1 = out + OFF_IDS1;
  float* ids2 = out + OFF_IDS2;
  float* st1  = out + OFF_ST1;
  float* st2  = out + OFF_ST2;
  unsigned* stg1g = stg1 + g * 512;       // group slice: 16 batches x 32 words

  for (int t = 0; t < SEQ; ++t) {
    // ================= Phase A : layer 1 (sparse gather, L2-fed) =========
    if (tid < 32) s_lm1[tid >> 1][tid & 1] = 0u;
    {
      const int* src = inp_ids + (size_t)(t * BATCH + g * 16) * SMAX;
      for (int i = tid; i < 16 * SMAX; i += THREADS) s_ids[i >> 6][i & 63] = src[i];
    }
    __syncthreads();

    const int num = inp_num[t * BATCH + b1];
    v4f acc = {0.f, 0.f, 0.f, 0.f};
    #pragma unroll 4
    for (int s = 0; s < num; ++s) {
      int id = s_ids[bl][s];
      acc += *(const v4f*)(W1T + (size_t)id * NN + n1);
    }
    unsigned nib = 0u; int cs = 0, ch = 0;
    for (int j = 0; j < 4; ++j) {
      float v = d1[j] * v1r[j] + (1.f - d1[j]) * acc[j];
      bool spk = v > t1[j];
      cs += spk; ch += (v > 0.9f * t1[j]);
      nib |= (spk ? 1u : 0u) << j;
      v1r[j] = spk ? 0.f : v;                              // reset-to-zero
    }
    int c0 = cs, c1 = ch;
    for (int o = 1; o < 16; o <<= 1) { c0 += __shfl_xor(c0, o, 32); c1 += __shfl_xor(c1, o, 32); }
    if (l16 == 0) {
      atomicAdd(&cnt1[(t * BATCH + b1) * 2 + 0], c0);
      atomicAdd(&cnt1[(t * BATCH + b1) * 2 + 1], c1);
    }
    { v4f o4 = {v1r[0], v1r[1], v1r[2], v1r[3]};
      *(v4f*)(st1 + (size_t)(t * BATCH + b1) * NN + n1) = o4; }
    atomicOr(&s_lm1[bl][l16 >> 3], nib << ((l16 & 7) * 4));
    __syncthreads();
    if (tid < 32) stg1g[(tid >> 1) * 32 + w * 2 + (tid & 1)] = s_lm1[tid >> 1][tid & 1];
    epoch = group_barrier(ctr, epoch);

    // ================= Phase B : layer 2 (f32 WMMA, LDS-fed) =============
    if (tid < 32) s_lm2[tid >> 1][tid & 1] = 0u;
    for (int i = tid; i < 512; i += THREADS) s_m1[i] = ld_dev(stg1g + i);
    __syncthreads();

    // C tile = [16 batches x 16 cols]; 4 tiles x 2 K-halves across 8 waves
    const int    khalf = wave >> 2;
    const int    ma    = lane & 15;         // A row (batch)
    const int    kb    = (lane >> 4) * 2;   // A/B K sub-lane offset
    const float* brow  = &s_w2[cl][0];
    v8f c8 = {0.f, 0.f, 0.f, 0.f, 0.f, 0.f, 0.f, 0.f};
    for (int wi = khalf * 16; wi < khalf * 16 + 16; ++wi) {
      const unsigned wdm   = s_m1[ma * 32 + wi];   // one mask word = 8 WMMAs
      const unsigned wd2   = wdm >> kb;            // hoist lane-variable shift
      const int      kbase = wi * 32;
      #pragma unroll
      for (int u = 0; u < 8; ++u) {
        // pair via constant-shift bfe; LUT at LDS 0 -> address = pair*8
        v2f a = *(const v2f*)&s_lut[((wd2 >> (u * 4)) & 3u) * 2];
        v2f b = *(const v2f*)&brow[kbase + u * 4 + kb];        // ds_load_b64
        c8 = __builtin_amdgcn_wmma_f32_16x16x4_f32(false, a, false, b,
                                                   (short)0, c8, false, false);
      }
    }
    if (wave >= 4)
      for (int r = 0; r < 8; ++r) s_acc[wave - 4][lane][r] = c8[r];
    __syncthreads();

    if (wave < 4) {
      for (int r = 0; r < 8; ++r) {
        float I2 = c8[r] + s_acc[wave][lane][r];
        float v  = d2 * v2r[r] + (1.f - d2) * I2;
        bool spk = v > t2;
        bool hi  = v > 0.9f * t2;
        unsigned long long bs = __ballot(spk);
        unsigned long long bh = __ballot(hi);
        v2r[r] = spk ? 0.f : v;
        st2[(size_t)(t * BATCH + g * 16 + r + mhi) * NN + col] = v2r[r];
        if (lane == 0) {
          unsigned lo = (unsigned)(bs & 0xFFFFu), lh = (unsigned)((bs >> 16) & 0xFFFFu);
          unsigned ho = (unsigned)(bh & 0xFFFFu), hh = (unsigned)((bh >> 16) & 0xFFFFu);
          atomicAdd(&cnt2[(t * BATCH + g * 16 + r) * 2 + 0],     __popc(lo));
          atomicAdd(&cnt2[(t * BATCH + g * 16 + r) * 2 + 1],     __popc(ho));
          atomicAdd(&cnt2[(t * BATCH + g * 16 + r + 8) * 2 + 0], __popc(lh));
          atomicAdd(&cnt2[(t * BATCH + g * 16 + r + 8) * 2 + 1], __popc(hh));
          unsigned shw = (unsigned)(tile & 1) * 16u;
          atomicOr(&s_lm2[r][tile >> 1],     lo << shw);
          atomicOr(&s_lm2[r + 8][tile >> 1], lh << shw);
        }
      }
    }
    __syncthreads();
    if (tid < 32)
      (stg2 + g * 512)[(tid >> 1) * 32 + w * 2 + (tid & 1)] = s_lm2[tid >> 1][tid & 1];
    epoch = group_barrier(ctr, epoch);

    // ================= Phase C : per-batch spike-id compaction ===========
    const int bsel = g * 16 + w;            // WG w owns batch w of its group
    if (tid < 32)
      emit_ids(stg1 + bsel * 32, ids1 + (size_t)(t * BATCH + bsel) * SMAX, tid);
    else if (tid < 64)
      emit_ids(stg2 + bsel * 32, ids2 + (size_t)(t * BATCH + bsel) * SMAX, tid - 32);
    epoch = group_barrier(ctr, epoch);      // staging reusable next step
  }
}

// ---------------------------------------------------------------------------
extern "C" void kernel_launch(void* const* d_in, const int* in_sizes, int n_in,
                              void* d_out, int out_size, void* d_ws, size_t ws_size,
                              hipStream_t stream) {
  const float* W1    = (const float*)d_in[0];
  const float* W2    = (const float*)d_in[1];
  const float* dec1  = (const float*)d_in[2];
  const float* dec2  = (const float*)d_in[3];
  const float* th1   = (const float*)d_in[4];
  const float* th2   = (const float*)d_in[5];
  const float* init0 = (const float*)d_in[6];
  const float* init1 = (const float*)d_in[7];
  const int*   ids   = (const int*)d_in[8];
  const int*   nums  = (const int*)d_in[9];
  float* out = (float*)d_out;
  char*  ws  = (char*)d_ws;

  float*    W1T = (float*)(ws + WS_W1T);
  unsigned* st1 = (unsigned*)(ws + WS_STG1);
  unsigned* st2 = (unsigned*)(ws + WS_STG2);
  unsigned* bar = (unsigned*)(ws + WS_BAR);

  const int ncnt = SEQ * BATCH * 4;                 // cnt1 + cnt2 (contiguous)
  init_k<<<(ncnt + 255) / 256, 256, 0, stream>>>((int*)out + OFF_CNT1, ncnt, bar, 64);

  dim3 tb(32, 8);
  transpose_k<<<1024, tb, 0, stream>>>(W1, W1T);    // W1T[in][n1]

  (void)hipFuncSetAttribute((const void*)lif_persistent,
                            hipFuncAttributeMaxDynamicSharedMemorySize,
                            (int)SMEM_BYTES);
  lif_persistent<<<GROUPS * WGPG, THREADS, (int)SMEM_BYTES, stream>>>(
      dec1, dec2, th1, th2, init0, init1, ids, nums, W1T, W2, st1, st2, bar, out);
}